// SinkhornImageLoss_30769145708888
// MI455X (gfx1250) — compile-verified
//
#include <hip/hip_runtime.h>
#include <math.h>

// ---------------- problem constants (match reference) ----------------
#define EPS      0.01f
#define INV_EPS  100.0f
#define N        2304          // H*W = 48*48
#define B        16
#define MAX_ITER 10

// ---------------- tiling ----------------
#define CHUNK    128           // columns staged per LDS chunk
#define CSTRIDE  132           // padded LDS row stride (floats) -> conflict-free lane=row reads
#define NCHUNKS  (N / CHUNK)   // 18
#define RPB      16            // rows per workgroup
#define NBLOCKS  (N / RPB)     // 144

typedef __attribute__((ext_vector_type(2))) float v2f;
typedef __attribute__((ext_vector_type(8))) float v8f;
typedef int v4i __attribute__((vector_size(16)));   // matches builtin's V4i

#define ASG __attribute__((address_space(1)))       // printed as __device__ by clang
#define ASL __attribute__((address_space(3)))       // printed as __shared__ by clang

// CDNA5 async copy-to-LDS path (ASYNCcnt), guarded so either toolchain compiles.
#if defined(__has_builtin)
#  if __has_builtin(__builtin_amdgcn_global_load_async_to_lds_b128) && \
      __has_builtin(__builtin_amdgcn_s_wait_asynccnt)
#    define USE_ASYNC_LDS 1
#  endif
#endif
#ifndef USE_ASYNC_LDS
#  define USE_ASYNC_LDS 0
#endif

// ============================================================================
// Prep: fu = eps*log_softmax(logits), ft = eps*log(target+1e-8), v = 0
// ============================================================================
__global__ void sink_prep(const float* __restrict__ logits,
                          const float* __restrict__ target,
                          float* __restrict__ fu,
                          float* __restrict__ ft,
                          float* __restrict__ v) {
  __shared__ float red[256];
  const int b = blockIdx.x;
  const int t = threadIdx.x;
  const float* x = logits + b * N;

  float m = -3.4e38f;
  for (int i = t; i < N; i += 256) m = fmaxf(m, x[i]);
  red[t] = m; __syncthreads();
  for (int s = 128; s > 0; s >>= 1) {
    if (t < s) red[t] = fmaxf(red[t], red[t + s]);
    __syncthreads();
  }
  m = red[0]; __syncthreads();

  float ssum = 0.f;
  for (int i = t; i < N; i += 256) ssum += __expf(x[i] - m);
  red[t] = ssum; __syncthreads();
  for (int s = 128; s > 0; s >>= 1) {
    if (t < s) red[t] += red[t + s];
    __syncthreads();
  }
  const float lse = m + __logf(red[0]);

  for (int i = t; i < N; i += 256) {
    fu[b * N + i] = EPS * (x[i] - lse);
    ft[b * N + i] = EPS * __logf(target[b * N + i] + 1e-8f);
    v[b * N + i]  = 0.0f;
  }
}

// ============================================================================
// Half-iteration: out[b,r] = logp[b,r] - eps*LSE_c( (w[b,c] - C[r,c]) / eps )
// 16 waves (wave <-> batch) x 16 rows per workgroup. C/w chunks staged to LDS
// via double-buffered GLOBAL_LOAD_ASYNC_TO_LDS_B128 (fallback: sync staging).
// Row sums of exp() computed with V_WMMA_F32_16X16X4_F32, B-matrix = ones.
// ============================================================================
__global__ void sink_lse(const float* __restrict__ C,
                         const float* __restrict__ w,
                         const float* __restrict__ logp,
                         float* __restrict__ out) {
#if USE_ASYNC_LDS
  __shared__ float Ct[2][RPB * CSTRIDE];
  __shared__ float Wt[2][B * CSTRIDE];
#else
  __shared__ float Ct[1][RPB * CSTRIDE];
  __shared__ float Wt[1][B * CSTRIDE];
#endif
  __shared__ float Slds[B * RPB];

  const int tid  = threadIdx.x;
  const int wave = tid >> 5;            // batch index
  const int lane = tid & 31;
  const int row  = lane & 15;           // WMMA A-layout: lane -> row
  const int coff = (lane >> 4) << 1;    // lanes 16..31 handle K+2,K+3
  const int r0   = blockIdx.x * RPB;

#if USE_ASYNC_LDS
  // one 16-byte quad per thread per buffer: thread -> (row tid>>5, 4 cols)
  const int srr = tid >> 5;
  const int scq = (tid & 31) << 2;
  #define STAGE_ASYNC(buf_, c0_)                                                 \
    do {                                                                         \
      __builtin_amdgcn_global_load_async_to_lds_b128(                            \
          (ASG v4i*)(C + (size_t)(r0 + srr) * N + (c0_) + scq),                  \
          (ASL v4i*)(&Ct[(buf_)][srr * CSTRIDE + scq]), 0, 0);                   \
      __builtin_amdgcn_global_load_async_to_lds_b128(                            \
          (ASG v4i*)(w + (size_t)srr * N + (c0_) + scq),                         \
          (ASL v4i*)(&Wt[(buf_)][srr * CSTRIDE + scq]), 0, 0);                   \
    } while (0)
#endif

  // ---------------- pass 1: global row max of x = (w - C)/eps ----------------
  float m = -3.4e38f;
#if USE_ASYNC_LDS
  STAGE_ASYNC(0, 0);
  for (int ch = 0; ch < NCHUNKS; ++ch) {
    if (ch + 1 < NCHUNKS) {
      STAGE_ASYNC((ch + 1) & 1, (ch + 1) * CHUNK);
      __builtin_amdgcn_s_wait_asynccnt(2);   // in-order: current buffer landed
    } else {
      __builtin_amdgcn_s_wait_asynccnt(0);
    }
    __syncthreads();
    const float* Cb = Ct[ch & 1];
    const float* Wb = Wt[ch & 1];
#pragma unroll 4
    for (int k0 = 0; k0 < CHUNK; k0 += 4) {
      const int c = k0 + coff;
      const float x0 = (Wb[wave * CSTRIDE + c]     - Cb[row * CSTRIDE + c])     * INV_EPS;
      const float x1 = (Wb[wave * CSTRIDE + c + 1] - Cb[row * CSTRIDE + c + 1]) * INV_EPS;
      m = fmaxf(m, fmaxf(x0, x1));
    }
    __syncthreads();                         // readers done before buffer reuse
  }
#else
  for (int ch = 0; ch < NCHUNKS; ++ch) {
    const int c0 = ch * CHUNK;
    __syncthreads();
    for (int t = tid; t < RPB * CHUNK; t += 512) {
      const int rr = t >> 7, cc = t & (CHUNK - 1);
      Ct[0][rr * CSTRIDE + cc] = C[(size_t)(r0 + rr) * N + c0 + cc];
      Wt[0][rr * CSTRIDE + cc] = w[rr * N + c0 + cc];
      if (ch + 1 < NCHUNKS)
        __builtin_prefetch(&C[(size_t)(r0 + rr) * N + c0 + CHUNK + cc], 0, 1);
    }
    __syncthreads();
#pragma unroll 4
    for (int k0 = 0; k0 < CHUNK; k0 += 4) {
      const int c = k0 + coff;
      const float x0 = (Wt[0][wave * CSTRIDE + c]     - Ct[0][row * CSTRIDE + c])     * INV_EPS;
      const float x1 = (Wt[0][wave * CSTRIDE + c + 1] - Ct[0][row * CSTRIDE + c + 1]) * INV_EPS;
      m = fmaxf(m, fmaxf(x0, x1));
    }
  }
#endif
  // combine the two half-wave column partitions of the same row
  m = fmaxf(m, __shfl_xor(m, 16, 32));

  // ---------------- pass 2: sum exp(x - m) via WMMA reduction ----------------
  v8f acc = {0.f, 0.f, 0.f, 0.f, 0.f, 0.f, 0.f, 0.f};
  const v2f bones = {1.0f, 1.0f};       // B matrix = ones -> D accumulates row sums
#if USE_ASYNC_LDS
  STAGE_ASYNC(0, 0);
  for (int ch = 0; ch < NCHUNKS; ++ch) {
    if (ch + 1 < NCHUNKS) {
      STAGE_ASYNC((ch + 1) & 1, (ch + 1) * CHUNK);
      __builtin_amdgcn_s_wait_asynccnt(2);
    } else {
      __builtin_amdgcn_s_wait_asynccnt(0);
    }
    __syncthreads();
    const float* Cb = Ct[ch & 1];
    const float* Wb = Wt[ch & 1];
#pragma unroll 4
    for (int k0 = 0; k0 < CHUNK; k0 += 4) {
      const int c = k0 + coff;
      const float x0 = (Wb[wave * CSTRIDE + c]     - Cb[row * CSTRIDE + c])     * INV_EPS;
      const float x1 = (Wb[wave * CSTRIDE + c + 1] - Cb[row * CSTRIDE + c + 1]) * INV_EPS;
      v2f a;
      a[0] = __expf(x0 - m);
      a[1] = __expf(x1 - m);
      acc = __builtin_amdgcn_wmma_f32_16x16x4_f32(false, a, false, bones,
                                                  (short)0, acc, false, false);
    }
    __syncthreads();
  }
#else
  for (int ch = 0; ch < NCHUNKS; ++ch) {
    const int c0 = ch * CHUNK;
    __syncthreads();
    for (int t = tid; t < RPB * CHUNK; t += 512) {
      const int rr = t >> 7, cc = t & (CHUNK - 1);
      Ct[0][rr * CSTRIDE + cc] = C[(size_t)(r0 + rr) * N + c0 + cc];
      Wt[0][rr * CSTRIDE + cc] = w[rr * N + c0 + cc];
    }
    __syncthreads();
#pragma unroll 4
    for (int k0 = 0; k0 < CHUNK; k0 += 4) {
      const int c = k0 + coff;
      const float x0 = (Wt[0][wave * CSTRIDE + c]     - Ct[0][row * CSTRIDE + c])     * INV_EPS;
      const float x1 = (Wt[0][wave * CSTRIDE + c + 1] - Ct[0][row * CSTRIDE + c + 1]) * INV_EPS;
      v2f a;
      a[0] = __expf(x0 - m);
      a[1] = __expf(x1 - m);
      acc = __builtin_amdgcn_wmma_f32_16x16x4_f32(false, a, false, bones,
                                                  (short)0, acc, false, false);
    }
  }
#endif

  // D layout: VGPR v holds row (v + 8*(lane>=16)); column N=0 lives in lane 0/16
  if (lane == 0) {
#pragma unroll
    for (int vv = 0; vv < 8; ++vv) Slds[wave * 16 + vv] = acc[vv];
  }
  if (lane == 16) {
#pragma unroll
    for (int vv = 0; vv < 8; ++vv) Slds[wave * 16 + 8 + vv] = acc[vv];
  }
  __syncthreads();
  if (lane < 16) {
    const float S = Slds[wave * 16 + lane];           // row 'lane' sum
    out[wave * N + r0 + lane] =
        logp[wave * N + r0 + lane] - EPS * m - EPS * __logf(S);
  }
}

// ============================================================================
// Final cost: mean_b sum_ij exp((u_i + v_j - C_ij)/eps) * C_ij
// ============================================================================
__global__ void sink_zero(float* __restrict__ out) {
  if (threadIdx.x == 0) out[0] = 0.0f;
}

__global__ void sink_cost(const float* __restrict__ C,
                          const float* __restrict__ u,
                          const float* __restrict__ v,
                          float* __restrict__ out) {
  const int tid  = threadIdx.x;
  const int b    = tid >> 5;            // wave <-> batch
  const int lane = tid & 31;
  const int r0   = blockIdx.x * RPB;

  float acc = 0.f;
  for (int r = 0; r < RPB; ++r) {
    const float ur = u[b * N + r0 + r];
    const float* Cr = C + (size_t)(r0 + r) * N;
    for (int c = lane; c < N; c += 32) {
      const float Cv = Cr[c];
      acc += __expf((ur + v[b * N + c] - Cv) * INV_EPS) * Cv;
    }
  }
#pragma unroll
  for (int off = 16; off >= 1; off >>= 1) acc += __shfl_xor(acc, off, 32);
  if (lane == 0) atomicAdd(out, acc * (1.0f / (float)B));
}

// ============================================================================
extern "C" void kernel_launch(void* const* d_in, const int* in_sizes, int n_in,
                              void* d_out, int out_size, void* d_ws, size_t ws_size,
                              hipStream_t stream) {
  (void)in_sizes; (void)n_in; (void)out_size; (void)ws_size;
  const float* logits = (const float*)d_in[0];   // (B,H,W)
  const float* target = (const float*)d_in[1];   // (B,H,W)
  const float* C      = (const float*)d_in[2];   // (1,N,N) symmetric
  float* out = (float*)d_out;

  float* u  = (float*)d_ws;          // B*N
  float* v  = u  + B * N;            // B*N
  float* fu = v  + B * N;            // B*N : eps*log_softmax(pred)
  float* ft = fu + B * N;            // B*N : eps*log(target+1e-8)

  sink_prep<<<B, 256, 0, stream>>>(logits, target, fu, ft, v);
  for (int it = 0; it < MAX_ITER; ++it) {
    sink_lse<<<NBLOCKS, 512, 0, stream>>>(C, v, fu, u);  // u-update (uses v)
    sink_lse<<<NBLOCKS, 512, 0, stream>>>(C, u, ft, v);  // v-update (uses new u, C symmetric)
  }
  sink_zero<<<1, 64, 0, stream>>>(out);
  sink_cost<<<NBLOCKS, 512, 0, stream>>>(C, u, v, out);
}